// surfacePropLoss_85667417686241
// MI455X (gfx1250) — compile-verified
//
#include <hip/hip_runtime.h>
#include <math.h>

typedef __attribute__((ext_vector_type(2))) float v2f;
typedef __attribute__((ext_vector_type(8))) float v8f;

#define NPTS 8192
#define NB 2
#define QTOT (NB * NPTS)        // 16384
#define KNN 16
#define PATCH_PTS 256
#define COS_THR (-3.6732051e-06f)   // cos(1.5708)
#define BIGF 1e30f

// D = A(16x4 f32) * B(4x16 f32) + C.  K=4 (xyz padded with 0).
__device__ inline v8f wmma_dot_tile(v2f a, v2f b) {
  v8f c = {0.f, 0.f, 0.f, 0.f, 0.f, 0.f, 0.f, 0.f};
  return __builtin_amdgcn_wmma_f32_16x16x4_f32(
      /*neg_a=*/false, a, /*neg_b=*/false, b,
      /*c_mod=*/(short)0, c, /*reuse_a=*/false, /*reuse_b=*/false);
}

// ---------------------------------------------------------------------------
// Squared norms of pointCloud and gtPoints.
// ---------------------------------------------------------------------------
__global__ void norms_kernel(const float* __restrict__ pc,
                             const float* __restrict__ gp,
                             float* __restrict__ pn2,
                             float* __restrict__ gn2) {
  int i = blockIdx.x * blockDim.x + threadIdx.x;
  if (i < QTOT) {
    float x = pc[i * 3 + 0], y = pc[i * 3 + 1], z = pc[i * 3 + 2];
    pn2[i] = x * x + y * y + z * z;
  } else if (i < 2 * QTOT) {
    int j = i - QTOT;
    float x = gp[j * 3 + 0], y = gp[j * 3 + 1], z = gp[j * 3 + 2];
    gn2[j] = x * x + y * y + z * z;
  }
}

// ---------------------------------------------------------------------------
// Nearest GT point per query (argmin of |g|^2 - 2 p.g), gather gtNormal.
// One wave (32 lanes) per 16-query tile; WMMA produces 16x16 dot tile.
// ---------------------------------------------------------------------------
__global__ __launch_bounds__(32) void nearest_gt_kernel(
    const float* __restrict__ pc, const float* __restrict__ gp,
    const float* __restrict__ gn2, const float* __restrict__ gtn,
    float* __restrict__ ideal) {
  __shared__ float sd[32];
  __shared__ int si[32];
  int lane = threadIdx.x;
  int n = lane & 15;
  int hi = lane >> 4;
  int qbase = blockIdx.x * 16;
  int batch = qbase / NPTS;
  int q = qbase + n;

  float qx = pc[q * 3 + 0], qy = pc[q * 3 + 1], qz = pc[q * 3 + 2];
  v2f bq;
  bq.x = hi ? qz : qx;       // B: K rows striped lo/hi half-wave
  bq.y = hi ? 0.f : qy;

  float bd = BIGF;
  int bidx = 0;
  int cand0 = batch * NPTS;
  for (int ct = 0; ct < NPTS / 16; ++ct) {
    int cbase = cand0 + ct * 16;
    int cm = cbase + n;
    float cx = gp[cm * 3 + 0], cy = gp[cm * 3 + 1], cz = gp[cm * 3 + 2];
    v2f ac;
    ac.x = hi ? cz : cx;      // A: M=lane&15, K split lo/hi half-wave
    ac.y = hi ? 0.f : cy;
    v8f c = wmma_dot_tile(ac, bq);
#pragma unroll
    for (int r = 0; r < 8; ++r) {
      int m = cbase + r + 8 * hi;          // C layout: M = r + 8*hi, N = lane&15
      float d = gn2[m] - 2.0f * c[r];
      if (d < bd) { bd = d; bidx = m; }
    }
  }
  sd[lane] = bd;
  si[lane] = bidx;
  __syncthreads();
  if (lane < 16) {
    float d0 = sd[lane], d1 = sd[lane + 16];
    int ii = (d1 < d0) ? si[lane + 16] : si[lane];
    ideal[q * 3 + 0] = gtn[ii * 3 + 0];
    ideal[q * 3 + 1] = gtn[ii * 3 + 1];
    ideal[q * 3 + 2] = gtn[ii * 3 + 2];
  }
}

// ---------------------------------------------------------------------------
// Top-16 KNN per query.  useMask: drop candidates with ideal_i.ideal_j <= cos.
// patchMode: candidates restricted to the 256-point patch of the query.
// Distance order key: |p_m|^2 - 2 p_m.q  (query norm constant -> dropped).
// Per-lane insertion-sorted top-16 over its half of each WMMA tile; final
// 2-way merge of lane n / lane n+16 lists through LDS.
// ---------------------------------------------------------------------------
__global__ __launch_bounds__(32) void knn_topk_kernel(
    const float* __restrict__ pts, const float* __restrict__ cn2,
    const float* __restrict__ ideal, int useMask, int patchMode,
    int* __restrict__ knnOut) {
  __shared__ float lv[32][KNN];
  __shared__ int li[32][KNN];
  int lane = threadIdx.x;
  int n = lane & 15;
  int hi = lane >> 4;
  int qbase = blockIdx.x * 16;
  int batch = qbase / NPTS;
  int q = qbase + n;

  float qx = pts[q * 3 + 0], qy = pts[q * 3 + 1], qz = pts[q * 3 + 2];
  v2f bq;
  bq.x = hi ? qz : qx;
  bq.y = hi ? 0.f : qy;
  v2f biq;
  biq.x = 0.f;
  biq.y = 0.f;
  if (useMask) {
    float ix = ideal[q * 3 + 0], iy = ideal[q * 3 + 1], iz = ideal[q * 3 + 2];
    biq.x = hi ? iz : ix;
    biq.y = hi ? 0.f : iy;
  }

  int candBase, candTiles;
  if (patchMode) {
    int patch = (qbase - batch * NPTS) / PATCH_PTS;
    candBase = batch * NPTS + patch * PATCH_PTS;
    candTiles = PATCH_PTS / 16;
  } else {
    candBase = batch * NPTS;
    candTiles = NPTS / 16;
  }

  float bv[KNN];
  int bi[KNN];
#pragma unroll
  for (int t = 0; t < KNN; ++t) { bv[t] = BIGF; bi[t] = 0; }

  for (int ct = 0; ct < candTiles; ++ct) {
    int cbase = candBase + ct * 16;
    int cm = cbase + n;
    float cx = pts[cm * 3 + 0], cy = pts[cm * 3 + 1], cz = pts[cm * 3 + 2];
    v2f ac;
    ac.x = hi ? cz : cx;
    ac.y = hi ? 0.f : cy;
    v8f dc = wmma_dot_tile(ac, bq);         // p_m . q_n tile
    v8f an = {0.f, 0.f, 0.f, 0.f, 0.f, 0.f, 0.f, 0.f};
    if (useMask) {
      float ix = ideal[cm * 3 + 0], iy = ideal[cm * 3 + 1],
            iz = ideal[cm * 3 + 2];
      v2f ai;
      ai.x = hi ? iz : ix;
      ai.y = hi ? 0.f : iy;
      an = wmma_dot_tile(ai, biq);          // ideal_m . ideal_n tile
    }
#pragma unroll
    for (int r = 0; r < 8; ++r) {
      int m = cbase + r + 8 * hi;
      float d = cn2[m] - 2.0f * dc[r];
      if (useMask && an[r] <= COS_THR) d = BIGF;
      if (d < bv[KNN - 1]) {
        bv[KNN - 1] = d;
        bi[KNN - 1] = m;
#pragma unroll
        for (int s = KNN - 1; s > 0; --s) {
          if (bv[s] < bv[s - 1]) {
            float tv = bv[s]; bv[s] = bv[s - 1]; bv[s - 1] = tv;
            int ti = bi[s]; bi[s] = bi[s - 1]; bi[s - 1] = ti;
          }
        }
      }
    }
  }

#pragma unroll
  for (int t = 0; t < KNN; ++t) { lv[lane][t] = bv[t]; li[lane][t] = bi[t]; }
  __syncthreads();
  if (lane < 16) {
    int ia = 0, ib = 0;
#pragma unroll
    for (int t = 0; t < KNN; ++t) {
      float va = lv[lane][ia];
      float vb = lv[lane + 16][ib];
      int sel;
      if (va <= vb) { sel = li[lane][ia]; ++ia; }
      else          { sel = li[lane + 16][ib]; ++ib; }
      knnOut[q * KNN + t] = sel;
    }
  }
}

// ---------------------------------------------------------------------------
// 3x3 covariance of re-centered KNN + closed-form symmetric eigensolve.
// normal = eigenvector of smallest eigenvalue; surfvar = l0 / trace.
// ---------------------------------------------------------------------------
__global__ void cov_eig_kernel(const float* __restrict__ pts,
                               const int* __restrict__ knn,
                               float* __restrict__ nrmOut,
                               float* __restrict__ svOut) {
  int q = blockIdx.x * blockDim.x + threadIdx.x;
  if (q >= QTOT) return;
  float px = pts[q * 3 + 0], py = pts[q * 3 + 1], pz = pts[q * 3 + 2];
  float a00 = 0.f, a01 = 0.f, a02 = 0.f, a11 = 0.f, a12 = 0.f, a22 = 0.f;
#pragma unroll
  for (int t = 0; t < KNN; ++t) {
    int j = knn[q * KNN + t];
    float vx = pts[j * 3 + 0] - px;
    float vy = pts[j * 3 + 1] - py;
    float vz = pts[j * 3 + 2] - pz;
    a00 += vx * vx; a01 += vx * vy; a02 += vx * vz;
    a11 += vy * vy; a12 += vy * vz; a22 += vz * vz;
  }
  float tr = a00 + a11 + a22;
  float qm = tr * (1.0f / 3.0f);
  float p1 = a01 * a01 + a02 * a02 + a12 * a12;
  float b00 = a00 - qm, b11 = a11 - qm, b22 = a22 - qm;
  float p2 = b00 * b00 + b11 * b11 + b22 * b22 + 2.0f * p1;
  float p = sqrtf(p2 * (1.0f / 6.0f) + 1e-30f);
  float inv = 1.0f / p;
  float c00 = b00 * inv, c11 = b11 * inv, c22 = b22 * inv;
  float c01 = a01 * inv, c02 = a02 * inv, c12 = a12 * inv;
  float detB = c00 * (c11 * c22 - c12 * c12)
             - c01 * (c01 * c22 - c12 * c02)
             + c02 * (c01 * c12 - c11 * c02);
  float r = 0.5f * detB;
  r = fminf(1.0f, fmaxf(-1.0f, r));
  float phi = acosf(r) * (1.0f / 3.0f);
  float e1 = qm + 2.0f * p * cosf(phi);                  // largest
  float e3 = qm + 2.0f * p * cosf(phi + 2.0943951f);     // smallest
  float l0 = e3;

  // null-space vector of (A - l0*I) via largest cross product of rows
  float r0x = a00 - l0, r0y = a01, r0z = a02;
  float r1x = a01, r1y = a11 - l0, r1z = a12;
  float r2x = a02, r2y = a12, r2z = a22 - l0;
  float u0x = r0y * r1z - r0z * r1y, u0y = r0z * r1x - r0x * r1z,
        u0z = r0x * r1y - r0y * r1x;
  float u1x = r0y * r2z - r0z * r2y, u1y = r0z * r2x - r0x * r2z,
        u1z = r0x * r2y - r0y * r2x;
  float u2x = r1y * r2z - r1z * r2y, u2y = r1z * r2x - r1x * r2z,
        u2z = r1x * r2y - r1y * r2x;
  float n0 = u0x * u0x + u0y * u0y + u0z * u0z;
  float n1 = u1x * u1x + u1y * u1y + u1z * u1z;
  float n2 = u2x * u2x + u2y * u2y + u2z * u2z;
  float vx = u0x, vy = u0y, vz = u0z, nn = n0;
  if (n1 > nn) { vx = u1x; vy = u1y; vz = u1z; nn = n1; }
  if (n2 > nn) { vx = u2x; vy = u2y; vz = u2z; nn = n2; }
  float rinv = rsqrtf(nn + 1e-30f);
  nrmOut[q * 3 + 0] = vx * rinv;
  nrmOut[q * 3 + 1] = vy * rinv;
  nrmOut[q * 3 + 2] = vz * rinv;
  svOut[q] = l0 / (tr + 1e-30f);
  (void)e1;
}

// ---------------------------------------------------------------------------
// Loss reductions (two-stage, deterministic).
// ---------------------------------------------------------------------------
__global__ void loss_partial_kernel(const float* __restrict__ np,
                                    const float* __restrict__ ng,
                                    const float* __restrict__ svp,
                                    const float* __restrict__ svg,
                                    float* __restrict__ partials) {
  __shared__ float s0[256];
  __shared__ float s1[256];
  int tid = threadIdx.x;
  float nl = 0.f, sl = 0.f;
  for (int i = blockIdx.x * blockDim.x + tid; i < QTOT;
       i += gridDim.x * blockDim.x) {
    float dx = fabsf(np[i * 3 + 0]) - fabsf(ng[i * 3 + 0]);
    float dy = fabsf(np[i * 3 + 1]) - fabsf(ng[i * 3 + 1]);
    float dz = fabsf(np[i * 3 + 2]) - fabsf(ng[i * 3 + 2]);
    nl += sqrtf(dx * dx + dy * dy + dz * dz);
    float ds = svp[i] - svg[i];
    sl += ds * ds;
  }
  s0[tid] = nl;
  s1[tid] = sl;
  __syncthreads();
  for (int o = 128; o > 0; o >>= 1) {
    if (tid < o) { s0[tid] += s0[tid + o]; s1[tid] += s1[tid + o]; }
    __syncthreads();
  }
  if (tid == 0) {
    partials[blockIdx.x * 2 + 0] = s0[0];
    partials[blockIdx.x * 2 + 1] = s1[0];
  }
}

__global__ void loss_final_kernel(const float* __restrict__ partials, int nb,
                                  float* __restrict__ out) {
  if (threadIdx.x == 0 && blockIdx.x == 0) {
    float a = 0.f, b = 0.f;
    for (int i = 0; i < nb; ++i) {
      a += partials[i * 2 + 0];
      b += partials[i * 2 + 1];
    }
    out[0] = a * (1.0f / (float)QTOT);   // W_NORMAL = 1
    out[1] = b * (1.0f / (float)QTOT);   // W_SURFVAR = 1
  }
}

// ---------------------------------------------------------------------------
extern "C" void kernel_launch(void* const* d_in, const int* in_sizes, int n_in,
                              void* d_out, int out_size, void* d_ws,
                              size_t ws_size, hipStream_t stream) {
  const float* pc = (const float*)d_in[0];   // pointCloud [2,8192,3]
  const float* gp = (const float*)d_in[1];   // gtPoints   [2,8192,3]
  const float* gtn = (const float*)d_in[2];  // gtNormal   [2,8192,3]
  float* out = (float*)d_out;                // [nl, sl, normal_global...]

  char* w = (char*)d_ws;
  float* pn2 = (float*)w;      w += QTOT * sizeof(float);
  float* gn2 = (float*)w;      w += QTOT * sizeof(float);
  float* ideal = (float*)w;    w += QTOT * 3 * sizeof(float);
  int* knng = (int*)w;         w += QTOT * KNN * sizeof(int);
  int* knnp = (int*)w;         w += QTOT * KNN * sizeof(int);
  float* npat = (float*)w;     w += QTOT * 3 * sizeof(float);
  float* svp = (float*)w;      w += QTOT * sizeof(float);
  float* svg = (float*)w;      w += QTOT * sizeof(float);
  float* partials = (float*)w; w += 64 * 2 * sizeof(float);
  float* ng_out = out + 2;     // normal_global goes straight into d_out

  norms_kernel<<<(2 * QTOT + 255) / 256, 256, 0, stream>>>(pc, gp, pn2, gn2);
  nearest_gt_kernel<<<QTOT / 16, 32, 0, stream>>>(pc, gp, gn2, gtn, ideal);
  knn_topk_kernel<<<QTOT / 16, 32, 0, stream>>>(pc, pn2, ideal, 1, 0, knng);
  knn_topk_kernel<<<QTOT / 16, 32, 0, stream>>>(pc, pn2, ideal, 0, 1, knnp);
  cov_eig_kernel<<<QTOT / 256, 256, 0, stream>>>(pc, knng, ng_out, svg);
  cov_eig_kernel<<<QTOT / 256, 256, 0, stream>>>(pc, knnp, npat, svp);
  loss_partial_kernel<<<64, 256, 0, stream>>>(npat, ng_out, svp, svg, partials);
  loss_final_kernel<<<1, 64, 0, stream>>>(partials, 64, out);
}